// LorentzMHA_39109972198036
// MI455X (gfx1250) — compile-verified
//
#include <hip/hip_runtime.h>
#include <math.h>

#define B_  2
#define L_  1024
#define D_  256
#define H_  8
#define N_  32
#define ND_ 33          // N+1 (hyperboloid dim)
#define DP_ 48          // padded dim stride (row-major q / pair-packed v)
#define KT_ 36          // padded dim count (dim-major k/v staging), 18 pairs
#define EPS_ 1e-6f

typedef float v2f __attribute__((ext_vector_type(2)));
typedef float v8f __attribute__((ext_vector_type(8)));

static __device__ __forceinline__ v8f wmma4(v2f a, v2f b, v8f c) {
  // D = A(16x4 f32) * B(4x16 f32) + C(16x16 f32)
  return __builtin_amdgcn_wmma_f32_16x16x4_f32(false, a, false, b, (short)0, c,
                                               false, false);
}

#define V8ZERO {0.f,0.f,0.f,0.f,0.f,0.f,0.f,0.f}

// ---------------------------------------------------------------------------
// Kernel 1: QKV projection + RoPE + hyperboloid lift; stage in WMMA layouts.
//  qf  : (BH, L, DP_)        row-major, comp 0 = -t  (Minkowski-flipped q)
//  vr2 : (BH, L/2, DP_, 2)   key-pair interleaved vh rows (tan B operand)
//  kT2 : (BH, KT_/2, L, 2)   dim-pair interleaved kh (scores B operand)
//  vT2 : (BH, KT_/2, L, 2)   dim-pair interleaved vh (alpha B operand)
// ---------------------------------------------------------------------------
__global__ __launch_bounds__(256) void k_qkv_rope_lift(
    const float* __restrict__ x,  const float* __restrict__ rc,
    const float* __restrict__ rs,
    const float* __restrict__ Wq, const float* __restrict__ bq,
    const float* __restrict__ Wk, const float* __restrict__ bk,
    const float* __restrict__ Wv, const float* __restrict__ bv,
    float* __restrict__ qf, float* __restrict__ vr2,
    float* __restrict__ kT2, float* __restrict__ vT2)
{
  const int lane = threadIdx.x & 31;
  const int wid  = threadIdx.x >> 5;
  const int l    = blockIdx.x * 8 + wid;       // global token row 0..2047
  const int b    = l >> 10;
  const int li   = l & 1023;

  // stage x row cooperatively (8 regs * 32 lanes = 256 values)
  float xr[8];
  #pragma unroll
  for (int t = 0; t < 8; ++t) xr[t] = x[(size_t)l * D_ + t * 32 + lane];

  float aq[H_], ak[H_], av[H_];
  #pragma unroll
  for (int h = 0; h < H_; ++h) { aq[h] = 0.f; ak[h] = 0.f; av[h] = 0.f; }

  #pragma unroll
  for (int t = 0; t < 8; ++t) {
    for (int s = 0; s < 32; ++s) {
      float xs = __shfl(xr[t], s, 32);
      int k = t * 32 + s;
      const float* wq = Wq + (size_t)k * (H_ * N_) + lane;
      const float* wk = Wk + (size_t)k * (H_ * N_) + lane;
      const float* wv = Wv + (size_t)k * (H_ * N_) + lane;
      #pragma unroll
      for (int h = 0; h < H_; ++h) {
        aq[h] = fmaf(xs, wq[h * N_], aq[h]);
        ak[h] = fmaf(xs, wk[h * N_], ak[h]);
        av[h] = fmaf(xs, wv[h * N_], av[h]);
      }
    }
  }

  const float cs  = rc[li * N_ + lane];
  const float sn  = rs[li * N_ + lane];
  const float sgn = (lane < 16) ? -1.f : 1.f;   // rot_half sign

  #pragma unroll
  for (int h = 0; h < H_; ++h) {
    float uq = aq[h] + bq[h * N_ + lane];
    float uk = ak[h] + bk[h * N_ + lane];
    float uv = av[h] + bv[h * N_ + lane];
    // RoPE: rot_half(u)[j] = (j<16 ? -u[j+16] : u[j-16])
    float rq = sgn * __shfl_xor(uq, 16, 32);
    float rk = sgn * __shfl_xor(uk, 16, 32);
    uq = uq * cs + rq * sn;
    uk = uk * cs + rk * sn;
    // lift: t = sqrt(1 + ||u||^2) (wave-wide reduction over 32 dims)
    float sq = uq * uq, sk = uk * uk, sv = uv * uv;
    #pragma unroll
    for (int m = 16; m >= 1; m >>= 1) {
      sq += __shfl_xor(sq, m, 32);
      sk += __shfl_xor(sk, m, 32);
      sv += __shfl_xor(sv, m, 32);
    }
    float tq = sqrtf(1.f + sq);
    float tk = sqrtf(1.f + sk);
    float tv = sqrtf(1.f + sv);

    const int bh = b * H_ + h;

    // q row-major (scores/alpha A operand + mu reload)
    float* qrow = qf + ((size_t)bh * L_ + li) * DP_;
    qrow[1 + lane] = uq;
    if (lane == 0) qrow[0] = -tq;
    if (lane < DP_ - ND_) qrow[ND_ + lane] = 0.f;

    // vh key-pair interleaved rows: elem(key,dim) at ((key>>1)*DP_+dim)*2+(key&1)
    float* vrow = vr2 + ((size_t)bh * (L_ / 2) + (li >> 1)) * DP_ * 2 + (li & 1);
    vrow[(1 + lane) * 2] = uv;
    if (lane == 0) vrow[0] = tv;
    if (lane < DP_ - ND_) vrow[(ND_ + lane) * 2] = 0.f;

    // kh/vh dim-pair interleaved: elem(dim,key) at ((dim>>1)*L + key)*2+(dim&1)
    float* kcol = kT2 + (size_t)bh * (KT_ / 2) * L_ * 2;
    float* vcol = vT2 + (size_t)bh * (KT_ / 2) * L_ * 2;
    {
      int d = 1 + lane;
      kcol[((size_t)(d >> 1) * L_ + li) * 2 + (d & 1)] = uk;
      vcol[((size_t)(d >> 1) * L_ + li) * 2 + (d & 1)] = uv;
    }
    if (lane == 0) { kcol[(size_t)li * 2] = tk; vcol[(size_t)li * 2] = tv; }
    if (lane < KT_ - ND_) {
      int d = ND_ + lane;
      kcol[((size_t)(d >> 1) * L_ + li) * 2 + (d & 1)] = 0.f;
      vcol[((size_t)(d >> 1) * L_ + li) * 2 + (d & 1)] = 0.f;
    }
  }
}

// ---------------------------------------------------------------------------
// Kernel 2: attention + Karcher step + origin log-map.
// One wave handles a 16-row q-tile of one (b,h); streams 64 key tiles twice.
// Softmax stats are kept per-lane; single butterfly merge after pass 1.
// ---------------------------------------------------------------------------
__global__ __launch_bounds__(32) void k_attn(
    const float* __restrict__ qf, const float* __restrict__ kT2,
    const float* __restrict__ vT2, const float* __restrict__ vr2,
    float* __restrict__ a_out, float* __restrict__ yb)
{
  __shared__ float lds_w[16 * 17];
  const int lane = threadIdx.x;
  const int half = lane >> 4;
  const int l16  = lane & 15;
  const int koff = half * 2;          // WMMA f32 K split across half-waves

  const int tile = blockIdx.x;
  const int qt = tile & 63;
  const int bh = tile >> 6;
  const int q0 = qt * 16;
  const int b  = bh >> 3;
  const int h  = bh & 7;

  const float* qfp  = qf  + ((size_t)bh * L_ + q0) * DP_;
  const float* kTp  = kT2 + (size_t)bh * (KT_ / 2) * L_ * 2;
  const float* vTp  = vT2 + (size_t)bh * (KT_ / 2) * L_ * 2;
  const float* vrp  = vr2 + (size_t)bh * (L_ / 2) * DP_ * 2;

  // A operand for q-tile: 16 rows x 36 dims = 9 K4-steps, kept in registers
  v2f aq[9];
  #pragma unroll
  for (int kk = 0; kk < 9; ++kk) {
    int d = 4 * kk + koff;                       // even
    aq[kk] = *(const v2f*)(qfp + l16 * DP_ + d);
  }

  // ---------------- pass 1: per-lane online softmax stats ----------------
  float mrow[8], srow[8];
  #pragma unroll
  for (int i = 0; i < 8; ++i) { mrow[i] = -3.0e38f; srow[i] = 0.f; }

  for (int j = 0; j < 64; ++j) {
    const int k0 = j * 16;
    if (j < 63) __builtin_prefetch(&kTp[(k0 + 16) * 2], 0, 0);  // next tile
    v8f c = V8ZERO;
    #pragma unroll
    for (int kk = 0; kk < 9; ++kk) {
      int d2 = (4 * kk + koff) >> 1;
      v2f bk_ = *(const v2f*)(kTp + ((size_t)d2 * L_ + k0 + l16) * 2);
      c = wmma4(aq[kk], bk_, c);
    }
    #pragma unroll
    for (int i = 0; i < 8; ++i) {          // per-lane update only
      float nm = fmaxf(mrow[i], c[i]);
      srow[i] = srow[i] * __expf(mrow[i] - nm) + __expf(c[i] - nm);
      mrow[i] = nm;
    }
  }
  // butterfly merge of (m, s) across the 16 key lanes (once)
  #pragma unroll
  for (int msk = 8; msk >= 1; msk >>= 1) {
    #pragma unroll
    for (int i = 0; i < 8; ++i) {
      float mo = __shfl_xor(mrow[i], msk, 32);
      float so = __shfl_xor(srow[i], msk, 32);
      float nm = fmaxf(mrow[i], mo);
      srow[i] = srow[i] * __expf(mrow[i] - nm) + so * __expf(mo - nm);
      mrow[i] = nm;
    }
  }
  float rsum[8];
  #pragma unroll
  for (int i = 0; i < 8; ++i) rsum[i] = 1.f / srow[i];

  // ---------------- pass 2: a, w, tangent accumulation ----------------
  v8f tc0 = V8ZERO, tc1 = V8ZERO, tc2 = V8ZERO;
  float wsa[8];
  #pragma unroll
  for (int i = 0; i < 8; ++i) wsa[i] = 0.f;

  float* arow = a_out + ((size_t)bh * L_ + q0) * L_;

  for (int j = 0; j < 64; ++j) {
    const int k0 = j * 16;
    v8f c  = V8ZERO;
    v8f al = V8ZERO;
    #pragma unroll
    for (int kk = 0; kk < 9; ++kk) {
      int d2 = (4 * kk + koff) >> 1;
      v2f bk_ = *(const v2f*)(kTp + ((size_t)d2 * L_ + k0 + l16) * 2);
      v2f bv_ = *(const v2f*)(vTp + ((size_t)d2 * L_ + k0 + l16) * 2);
      c  = wmma4(aq[kk], bk_, c);    // Lorentz scores
      al = wmma4(aq[kk], bv_, al);   // <mflip(q), v>  (alpha = -this)
    }
    #pragma unroll
    for (int i = 0; i < 8; ++i) {
      float alpha = fmaxf(-al[i], 1.f + EPS_);
      float coef  = acoshf(alpha) * rsqrtf(alpha * alpha - 1.f);
      float p     = __expf(c[i] - mrow[i]) * rsum[i];
      int   m     = i + 8 * half;
      arow[(size_t)m * L_ + k0 + l16] = p;          // attention output
      float w = p * coef;
      wsa[i] += w * alpha;                          // per-lane accumulate
      lds_w[m * 17 + l16] = w;                      // C-layout -> LDS
    }
    __syncthreads();
    // tan += w @ vh   (A = w from LDS in A-layout, B = vh key-pair packed)
    #pragma unroll
    for (int kk = 0; kk < 4; ++kk) {
      const int kidx = k0 + 4 * kk + koff;          // even key index
      v2f aw;
      aw.x = lds_w[l16 * 17 + 4 * kk + koff];
      aw.y = lds_w[l16 * 17 + 4 * kk + koff + 1];
      const float* pb = vrp + (size_t)(kidx >> 1) * DP_ * 2;
      v2f b0 = *(const v2f*)(pb + (l16)      * 2);
      v2f b1 = *(const v2f*)(pb + (16 + l16) * 2);
      v2f b2 = *(const v2f*)(pb + (32 + l16) * 2);
      tc0 = wmma4(aw, b0, tc0);
      tc1 = wmma4(aw, b1, tc1);
      tc2 = wmma4(aw, b2, tc2);
    }
    __syncthreads();
  }

  // reduce wsum(alpha) across the 16 key lanes
  #pragma unroll
  for (int i = 0; i < 8; ++i) {
    #pragma unroll
    for (int m = 8; m >= 1; m >>= 1) wsa[i] += __shfl_xor(wsa[i], m, 32);
  }

  // tan = (w@vh) - wsum*mu ;  mu = qh (un-flip component 0)
  float tanv[3][8], muv[3][8];
  #pragma unroll
  for (int sub = 0; sub < 3; ++sub) {
    #pragma unroll
    for (int i = 0; i < 8; ++i) {
      int m = i + 8 * half;
      int d = sub * 16 + l16;
      float qv = qfp[m * DP_ + d];
      float mu = (d == 0) ? -qv : qv;
      muv[sub][i] = mu;
      float tval = (sub == 0) ? tc0[i] : ((sub == 1) ? tc1[i] : tc2[i]);
      tanv[sub][i] = tval - wsa[i] * mu;
    }
  }
  // Lorentzian norm of tangent per row
  float vn[8];
  #pragma unroll
  for (int i = 0; i < 8; ++i) {
    float part = 0.f;
    #pragma unroll
    for (int sub = 0; sub < 3; ++sub) {
      float tt = tanv[sub][i] * tanv[sub][i];
      if (sub == 0 && l16 == 0) tt = -tt;   // metric diag(-1,1,...)
      part += tt;
    }
    #pragma unroll
    for (int m = 8; m >= 1; m >>= 1) part += __shfl_xor(part, m, 32);
    vn[i] = sqrtf(fmaxf(part, 1e-12f));
  }
  // exp-map update mu' = cosh(vn)*mu + sinh(vn)/vn * tan
  float mnew[3][8];
  #pragma unroll
  for (int i = 0; i < 8; ++i) {
    float ch = coshf(vn[i]);
    float sh = sinhf(vn[i]) / vn[i];
    #pragma unroll
    for (int sub = 0; sub < 3; ++sub)
      mnew[sub][i] = ch * muv[sub][i] + sh * tanv[sub][i];
  }
  // origin log-map and write y (B, L, H*33)
  #pragma unroll
  for (int i = 0; i < 8; ++i) {
    float t0 = __shfl(mnew[0][i], half * 16, 32);   // broadcast d==0 elem
    float t  = fmaxf(t0, 1.f + EPS_);
    float co = acoshf(t) * rsqrtf(t * t - 1.f);
    int   m  = i + 8 * half;
    float* yrow = yb + (size_t)(b * L_ + q0 + m) * (H_ * ND_) + h * ND_;
    #pragma unroll
    for (int sub = 0; sub < 3; ++sub) {
      int d = sub * 16 + l16;
      if (d < ND_) yrow[d] = co * (mnew[sub][i] - ((d == 0) ? t : 0.f));
    }
  }
}

// ---------------------------------------------------------------------------
// Pack Wo (264 x 256) into K-pair interleaved layout for b64 B-operand loads:
//   Wo2[((k>>1)*256 + n)*2 + (k&1)] = Wo[k*256 + n]
// ---------------------------------------------------------------------------
__global__ __launch_bounds__(256) void k_pack_wo(const float* __restrict__ Wo,
                                                 float* __restrict__ Wo2)
{
  const int k = blockIdx.x;            // 0..263
  const int n = threadIdx.x;           // 0..255
  Wo2[((size_t)(k >> 1) * D_ + n) * 2 + (k & 1)] = Wo[(size_t)k * D_ + n];
}

// ---------------------------------------------------------------------------
// Kernel 3: z = y @ Wo + bo    (2048 x 256 x 264, pure f32 WMMA)
// ---------------------------------------------------------------------------
__global__ __launch_bounds__(32) void k_outproj(
    const float* __restrict__ yb, const float* __restrict__ Wo2,
    const float* __restrict__ bo, float* __restrict__ z)
{
  const int lane = threadIdx.x;
  const int half = lane >> 4;
  const int l16  = lane & 15;
  const int koff = half * 2;
  const int m0 = blockIdx.x * 16;
  const int n0 = blockIdx.y * 16;
  const int KD = H_ * ND_;            // 264 = 66 exact K4 steps

  v8f c = V8ZERO;
  for (int k0 = 0; k0 < KD; k0 += 4) {
    v2f a  = *(const v2f*)(yb + (size_t)(m0 + l16) * KD + k0 + koff);
    v2f bb = *(const v2f*)(Wo2 + ((size_t)((k0 + koff) >> 1) * D_ + n0 + l16) * 2);
    c = wmma4(a, bb, c);
  }
  const float bias = bo[n0 + l16];
  #pragma unroll
  for (int i = 0; i < 8; ++i) {
    int m = m0 + i + 8 * half;
    z[(size_t)m * D_ + n0 + l16] = c[i] + bias;
  }
}

// ---------------------------------------------------------------------------
extern "C" void kernel_launch(void* const* d_in, const int* in_sizes, int n_in,
                              void* d_out, int out_size, void* d_ws, size_t ws_size,
                              hipStream_t stream) {
  const float* x   = (const float*)d_in[0];
  const float* rc  = (const float*)d_in[1];
  const float* rs  = (const float*)d_in[2];
  const float* Wq  = (const float*)d_in[3];
  const float* bq  = (const float*)d_in[4];
  const float* Wk  = (const float*)d_in[5];
  const float* bk  = (const float*)d_in[6];
  const float* Wv  = (const float*)d_in[7];
  const float* bv  = (const float*)d_in[8];
  const float* Wo  = (const float*)d_in[9];
  const float* bo  = (const float*)d_in[10];

  float* z_out = (float*)d_out;                         // (B, L, D)
  float* a_out = z_out + (size_t)B_ * L_ * D_;          // (B, H, L, L)

  // workspace layout (floats)
  float* ws = (float*)d_ws;
  const size_t n_qf = (size_t)B_ * H_ * L_ * DP_;       // 786432
  const size_t n_vr = (size_t)B_ * H_ * L_ * DP_;       // 786432 (pair-packed)
  const size_t n_kT = (size_t)B_ * H_ * KT_ * L_;       // 589824 (pair-packed)
  const size_t n_vT = (size_t)B_ * H_ * KT_ * L_;       // 589824 (pair-packed)
  const size_t n_yb = (size_t)B_ * L_ * (H_ * ND_);     // 540672
  float* qf  = ws;
  float* vr2 = qf  + n_qf;
  float* kT2 = vr2 + n_vr;
  float* vT2 = kT2 + n_kT;
  float* yb  = vT2 + n_vT;
  float* Wo2 = yb  + n_yb;                              // 67584

  k_qkv_rope_lift<<<(B_ * L_) / 8, 256, 0, stream>>>(
      x, rc, rs, Wq, bq, Wk, bk, Wv, bv, qf, vr2, kT2, vT2);

  k_pack_wo<<<H_ * ND_, D_, 0, stream>>>(Wo, Wo2);

  k_attn<<<B_ * H_ * (L_ / 16), 32, 0, stream>>>(qf, kT2, vT2, vr2, a_out, yb);

  k_outproj<<<dim3((B_ * L_) / 16, D_ / 16), 32, 0, stream>>>(yb, Wo2, bo, z_out);
}